// VectorQuantizer_55250459295889
// MI455X (gfx1250) — compile-verified
//
#include <hip/hip_runtime.h>
#include <hip/hip_fp16.h>

#define R_PTS   2048   // points per problem
#define KC      256    // clusters
#define NPROB   512    // sub-problems (columns/4)
#define THREADS 256
#define ITERS   10

typedef float v2f __attribute__((ext_vector_type(2)));
typedef float v8f __attribute__((ext_vector_type(8)));
typedef unsigned int u32x4 __attribute__((ext_vector_type(4)));
typedef int i32x4 __attribute__((ext_vector_type(4)));
typedef int i32x8 __attribute__((ext_vector_type(8)));

#ifndef __has_builtin
#define __has_builtin(x) 0
#endif

__device__ __forceinline__ float h16(float x) { return __half2float(__float2half(x)); }

// ---------------------------------------------------------------------------
// TDM: load a (tile_w x tile_h) fp32 tile (row stride = stride elems) from
// global into LDS via the Tensor Data Mover. 6-arg clang-23 builtin form:
//   (u32x4 g0, i32x8 g1, i32x4 g2, i32x4 g3, i32x8 extra, i32 cpol)
// D# group0: count=1 | lds_addr | global_addr[56:0] | type=2
// D# group1: data_size=4B, tensor_dim0/1 = 2048, tile_dim0=tile_w,
//            tile_dim1=tile_h, tensor_dim0_stride=stride
// ---------------------------------------------------------------------------
#if __has_builtin(__builtin_amdgcn_tensor_load_to_lds)
#define HAVE_TDM 1
__device__ __forceinline__ void tdm_load_tile_f32(const float* gsrc, void* lds_dst,
                                                  int tile_w, int tile_h, int stride)
{
    unsigned long long ga = (unsigned long long)(uintptr_t)gsrc;
    unsigned lds = (unsigned)(uintptr_t)lds_dst;   // low 32 bits of flat LDS ptr = LDS offset

    u32x4 g0;
    g0[0] = 1u;                                           // count=1 (valid), user mode
    g0[1] = lds;                                          // lds_addr
    g0[2] = (unsigned)(ga & 0xFFFFFFFFu);                 // global_addr[31:0]
    g0[3] = (unsigned)((ga >> 32) & 0x01FFFFFFu) | 0x80000000u; // addr[56:32] | type=2<<30

    i32x8 g1;
    g1[0] = 0x00020000;                  // workgroup_mask=0, data_size=2 (4 bytes)
    g1[1] = (int)(2048u << 16);          // tensor_dim0[15:0] in [31:16] -> 2048
    g1[2] = (int)(2048u << 16);          // tensor_dim0 hi=0 | tensor_dim1[15:0]=2048
    g1[3] = (int)((unsigned)tile_w << 16); // tensor_dim1 hi=0 | tile_dim0
    g1[4] = tile_h;                      // tile_dim1 | tile_dim2=0
    g1[5] = stride;                      // tensor_dim0_stride[31:0]
    g1[6] = 0;                           // stride hi | tensor_dim1_stride lo (unused 2D)
    g1[7] = 0;

    i32x4 z4 = {0, 0, 0, 0};
    i32x8 z8 = {0, 0, 0, 0, 0, 0, 0, 0};
    __builtin_amdgcn_tensor_load_to_lds(g0, g1, z4, z4, z8, 0);
#if __has_builtin(__builtin_amdgcn_s_wait_tensorcnt)
    __builtin_amdgcn_s_wait_tensorcnt(0);
#else
    asm volatile("s_wait_tensorcnt 0x0" ::: "memory");
#endif
}
#else
#define HAVE_TDM 0
#endif

// ---------------------------------------------------------------------------
// Stage 1: k-means++ sampled init. One workgroup (8 waves) per problem.
// Xg[p][r][:] = X[r][4p..4p+3]
// ---------------------------------------------------------------------------
__global__ __launch_bounds__(THREADS)
void kpp_init_kernel(const float* __restrict__ X,
                     const float* __restrict__ V,
                     __half* __restrict__ cent)
{
    __shared__ float4 spts[R_PTS];     // points (32 KB)
    __shared__ float  ssq[R_PTS];      // |x|^2 fp16-rounded (8 KB)
    __shared__ float  smind[R_PTS];    // current min distance (8 KB)
    __shared__ float  swsum[THREADS / 32];
    __shared__ int    sidx;

    const int p    = blockIdx.x;
    const int tid  = threadIdx.x;
    const int lane = tid & 31;
    const int wave = tid >> 5;

    // ---- stage X column-slice into LDS ----
#if HAVE_TDM
    if (tid == 0)
        tdm_load_tile_f32(X + (size_t)p * 4, &spts[0], /*w*/4, /*h*/R_PTS, /*stride*/2048);
#else
    {
        const float4* Xr = (const float4*)X;
        for (int r = tid; r < R_PTS; r += THREADS)
            spts[r] = Xr[(size_t)r * 512 + p];
    }
#endif
    __syncthreads();

    // fp16-round in place, compute |x|^2
    for (int r = tid; r < R_PTS; r += THREADS) {
        float4 x = spts[r];
        float4 hv = make_float4(h16(x.x), h16(x.y), h16(x.z), h16(x.w));
        spts[r] = hv;
        ssq[r]  = h16(hv.x*hv.x + hv.y*hv.y + hv.z*hv.z + hv.w*hv.w);
    }
    __syncthreads();

    const float4 c0  = spts[0];
    const float  sq0 = ssq[0];
    for (int r = tid; r < R_PTS; r += THREADS) {
        float4 q = spts[r];
        float d2 = ssq[r] + sq0 - 2.f*(q.x*c0.x + q.y*c0.y + q.z*c0.z + q.w*c0.w);
        smind[r] = h16(sqrtf(fmaxf(h16(d2), 0.f)));
    }
    if (tid == 0) {
        __half* cp = cent + ((size_t)p * KC) * 4;
        cp[0] = __float2half(c0.x); cp[1] = __float2half(c0.y);
        cp[2] = __float2half(c0.z); cp[3] = __float2half(c0.w);
    }
    __syncthreads();

    for (int i = 1; i < KC; i++) {
        const int base = tid * 8;
        float m8[8]; float loc = 0.f;
        #pragma unroll
        for (int j = 0; j < 8; j++) { m8[j] = smind[base + j]; loc += m8[j]; }

        // wave-level inclusive scan of per-thread sums (branchless shfl_up)
        float s = loc;
        #pragma unroll
        for (int off = 1; off < 32; off <<= 1) {
            float t = __shfl_up(s, off, 32);
            s += (lane >= off) ? t : 0.f;
        }
        if (lane == 31) swsum[wave] = s;
        if (tid == 0) sidx = R_PTS - 1;     // searchsorted clip to R-1
        __syncthreads();

        float waveoff = 0.f, total = 0.f;
        #pragma unroll
        for (int w = 0; w < THREADS / 32; w++) {
            float ws = swsum[w];
            waveoff += (w < wave) ? ws : 0.f;
            total   += ws;
        }
        const float target = V[(size_t)p * KC + i] * total;

        // first r with inclusive-cumsum(mind)[r] >= vi * total
        float run = waveoff + (s - loc);    // exclusive prefix of this thread
        int cand = -1;
        #pragma unroll
        for (int j = 0; j < 8; j++) {
            run += m8[j];
            cand = ((cand < 0) & (run >= target)) ? (base + j) : cand;
        }
        if (cand >= 0) atomicMin(&sidx, cand);   // ds_min_i32
        __syncthreads();

        const int idx = sidx;
        const float4 pc  = spts[idx];
        const float  sqc = ssq[idx];
        if (tid == 0) {
            __half* cp = cent + ((size_t)p * KC + i) * 4;
            cp[0] = __float2half(pc.x); cp[1] = __float2half(pc.y);
            cp[2] = __float2half(pc.z); cp[3] = __float2half(pc.w);
        }
        for (int r = tid; r < R_PTS; r += THREADS) {
            float4 q = spts[r];
            float d2 = ssq[r] + sqc - 2.f*(q.x*pc.x + q.y*pc.y + q.z*pc.z + q.w*pc.w);
            float d = h16(sqrtf(fmaxf(h16(d2), 0.f)));
            smind[r] = fminf(smind[r], d);
        }
        __syncthreads();
    }
}

// ---------------------------------------------------------------------------
// Stage 2: one Lloyd iteration. One workgroup per problem, 8 waves.
// dist_eff = |c|^2 - 2 x.c  via V_WMMA_F32_16X16X4_F32 (K=4 == SUB_VECTOR)
// All 16 B fragments (-2*C^T tiles) + |c|^2 biases are hoisted to registers.
// ---------------------------------------------------------------------------
__global__ __launch_bounds__(THREADS)
void kmeans_iter_kernel(const float* __restrict__ X,
                        __half* __restrict__ cent)
{
    __shared__ float4 spts[R_PTS];     // raw fp32 points (32 KB)
    __shared__ float  scb[KC * 4];     // -2 * centroid   (4 KB)
    __shared__ float  scn[KC];         // |c|^2           (1 KB)
    __shared__ float  sacc[KC * 5];    // sums(4) + count (5 KB)

    const int p   = blockIdx.x;
    const int tid = threadIdx.x;

    {   // centroids fp16 -> fp32, build B operand + bias, zero accumulators
        const __half* cp = cent + ((size_t)p * KC + tid) * 4;
        float c0 = __half2float(cp[0]), c1 = __half2float(cp[1]);
        float c2 = __half2float(cp[2]), c3 = __half2float(cp[3]);
        scb[tid*4+0] = -2.f*c0; scb[tid*4+1] = -2.f*c1;
        scb[tid*4+2] = -2.f*c2; scb[tid*4+3] = -2.f*c3;
        scn[tid] = c0*c0 + c1*c1 + c2*c2 + c3*c3;
        #pragma unroll
        for (int d = 0; d < 5; d++) sacc[tid*5 + d] = 0.f;
    }
#if HAVE_TDM
    if (tid == 0)
        tdm_load_tile_f32(X + (size_t)p * 4, &spts[0], 4, R_PTS, 2048);
#else
    {
        const float4* Xr = (const float4*)X;
        for (int r = tid; r < R_PTS; r += THREADS)
            spts[r] = Xr[(size_t)r * 512 + p];
    }
#endif
    __syncthreads();

    const int wave   = tid >> 5;
    const int lane   = tid & 31;
    const int halfId = lane >> 4;     // 0: K{0,1} / D rows M 0..7; 1: K{2,3} / M 8..15
    const int myN    = lane & 15;
    const float2* spts2 = (const float2*)spts;
    const float2* scb2  = (const float2*)scb;

    // hoist all 16 B fragments + biases into registers (invariant over m-tiles)
    v2f   bfrag[KC / 16];
    float cnv[KC / 16];
    #pragma unroll
    for (int nt = 0; nt < KC / 16; nt++) {
        const int n = nt * 16 + myN;
        float2 bb = scb2[n * 2 + halfId];
        bfrag[nt][0] = bb.x; bfrag[nt][1] = bb.y;
        cnv[nt] = scn[n];
    }

    for (int mt = wave; mt < R_PTS / 16; mt += THREADS / 32) {   // 16 tiles/wave, uniform
        // A fragment (16x4 f32): lanes 0-15 K={0,1}, lanes 16-31 K={2,3}, M = lane&15
        const int mA = mt * 16 + myN;
        float2 av = spts2[mA * 2 + halfId];
        v2f a; a[0] = av.x; a[1] = av.y;

        float minv[8]; int mink[8];
        #pragma unroll
        for (int q = 0; q < 8; q++) { minv[q] = 3.4e38f; mink[q] = 0; }

        #pragma unroll
        for (int nt = 0; nt < KC / 16; nt++) {
            const int n = nt * 16 + myN;
            v8f c;
            #pragma unroll
            for (int q = 0; q < 8; q++) c[q] = cnv[nt];

            v8f dmat = __builtin_amdgcn_wmma_f32_16x16x4_f32(
                false, a, false, bfrag[nt], (short)0, c, false, false);

            #pragma unroll
            for (int q = 0; q < 8; q++) {            // branchless running argmin
                float dv = dmat[q];
                int lt = dv < minv[q];
                minv[q] = lt ? dv : minv[q];
                mink[q] = lt ? n  : mink[q];
            }
        }
        // branchless argmin reduce across the 16 lanes of each half
        #pragma unroll
        for (int off = 1; off < 16; off <<= 1) {
            #pragma unroll
            for (int q = 0; q < 8; q++) {
                float ov = __shfl_xor(minv[q], off, 32);
                int   oi = __shfl_xor(mink[q], off, 32);
                int better = (int)(ov < minv[q]) |
                             ((int)(ov == minv[q]) & (int)(oi < mink[q]));
                minv[q] = better ? ov : minv[q];
                mink[q] = better ? oi : mink[q];
            }
        }
        // accumulate: D row M = q + 8*halfId; lanes 0..3 add dims, lane 4 adds count
        #pragma unroll
        for (int q = 0; q < 8; q++) {
            const int m = mt * 16 + q + halfId * 8;
            const int k = mink[q];
            if (myN < 4) {
                const float* xp = (const float*)&spts[m];
                atomicAdd(&sacc[k * 5 + myN], xp[myN]);     // ds_add_f32
            } else if (myN == 4) {
                atomicAdd(&sacc[k * 5 + 4], 1.0f);
            }
        }
    }
    __syncthreads();

    {   // m-step: mean with count clipped at 1, round to fp16
        const float cntv = sacc[tid*5 + 4];
        const float inv  = 1.f / fmaxf(cntv, 1.f);
        __half* cp = cent + ((size_t)p * KC + tid) * 4;
        cp[0] = __float2half(sacc[tid*5 + 0] * inv);
        cp[1] = __float2half(sacc[tid*5 + 1] * inv);
        cp[2] = __float2half(sacc[tid*5 + 2] * inv);
        cp[3] = __float2half(sacc[tid*5 + 3] * inv);
    }
}

// ---------------------------------------------------------------------------
// Stage 3: fp16 centroids -> fp32 output
// ---------------------------------------------------------------------------
__global__ void finalize_kernel(const __half* __restrict__ cent,
                                float* __restrict__ out, int n)
{
    int i = blockIdx.x * blockDim.x + threadIdx.x;
    if (i < n) out[i] = __half2float(cent[i]);
}

extern "C" void kernel_launch(void* const* d_in, const int* in_sizes, int n_in,
                              void* d_out, int out_size, void* d_ws, size_t ws_size,
                              hipStream_t stream) {
    (void)in_sizes; (void)n_in; (void)ws_size;
    const float* X = (const float*)d_in[0];   // (2048, 2048) f32
    const float* V = (const float*)d_in[1];   // (512, 256)  f32
    float* out = (float*)d_out;               // (512, 256, 4) f32
    __half* cent = (__half*)d_ws;             // (512, 256, 4) f16 scratch (1 MB)

    kpp_init_kernel<<<NPROB, THREADS, 0, stream>>>(X, V, cent);
    for (int it = 0; it < ITERS; ++it)
        kmeans_iter_kernel<<<NPROB, THREADS, 0, stream>>>(X, cent);

    const int n = NPROB * KC * 4;
    finalize_kernel<<<(n + THREADS - 1) / THREADS, THREADS, 0, stream>>>(cent, out, (n < out_size) ? n : out_size);
}